// RGCN_65025804861440
// MI455X (gfx1250) — compile-verified
//
#include <hip/hip_runtime.h>

#define N_NODES 50000
#define N_EDGES 1600000
#define CH 128
#define NREL 8

typedef __attribute__((ext_vector_type(2))) float v2f;
typedef __attribute__((ext_vector_type(8))) float v8f;

// ---------------------------------------------------------------------------
// Zero a float buffer (float4 stores, exact coverage)
// ---------------------------------------------------------------------------
__global__ void zero_f32(float4* __restrict__ p, int n4) {
  int i = blockIdx.x * 256 + threadIdx.x;
  if (i < n4) p[i] = make_float4(0.f, 0.f, 0.f, 0.f);
}

// ---------------------------------------------------------------------------
// Edge scatter for one relation: tmp[dst] += h[src] for edges with et==rel.
// One wave (32 lanes) per edge; float4 coalesced gather + 4 f32 atomics/lane.
// ---------------------------------------------------------------------------
__global__ void rgcn_scatter(const float* __restrict__ h,
                             const int* __restrict__ src,
                             const int* __restrict__ dst,
                             const int* __restrict__ et,
                             int rel, float* __restrict__ tmp) {
  int gid  = blockIdx.x * blockDim.x + threadIdx.x;
  int e    = gid >> 5;
  int lane = gid & 31;
  if (e >= N_EDGES) return;
  if (et[e] != rel) return;
  int s = src[e];
  int d = dst[e];
  const float4* hp = (const float4*)(h + (size_t)s * CH);
  float4 v = hp[lane];
  float* o = tmp + (size_t)d * CH + lane * 4;
  atomicAdd(o + 0, v.x);
  atomicAdd(o + 1, v.y);
  atomicAdd(o + 2, v.z);
  atomicAdd(o + 3, v.w);
}

// ---------------------------------------------------------------------------
// WMMA fp32 GEMM: D = A[nRows,128] @ B[128,128] (+ Cin) (+ resid) (relu) (+bias)
// Block = 256 threads = 8 waves. Weights staged in LDS in k-pair-interleaved
// layout: Bs[k/2][col] = {B[k][col], B[k+1][col]} so every B fragment is one
// aligned ds_load_b64 straight into an even VGPR pair for WMMA (no v_movs).
// Each wave computes a 16x128 output strip: 32 k-steps x 8 col tiles of
// V_WMMA_F32_16X16X4_F32, reusing the A fragment across the 8 tiles.
// ---------------------------------------------------------------------------
template <bool ACC, bool RELU, bool RESID, bool BIAS>
__global__ void __launch_bounds__(256)
rgcn_gemm(const float* __restrict__ A, const float* __restrict__ B,
          const float* __restrict__ Cin, float* __restrict__ D,
          const float* __restrict__ resid, const float* __restrict__ bias,
          int nRows) {
  __shared__ float2 Bs[(CH / 2) * CH];     // 64 KB of the 320 KB WGP LDS
  for (int i = threadIdx.x; i < (CH / 2) * CH; i += 256) {
    int k2  = i >> 7;                      // k pair index (k = 2*k2)
    int col = i & (CH - 1);
    float2 t;
    t.x = B[(size_t)(2 * k2) * CH + col];
    t.y = B[(size_t)(2 * k2 + 1) * CH + col];
    Bs[i] = t;
  }
  __syncthreads();

  int wave  = threadIdx.x >> 5;
  int lane  = threadIdx.x & 31;
  int strip = blockIdx.x * 8 + wave;
  int m0    = strip * 16;
  if (m0 >= nRows) return;

  int half = lane >> 4;                    // 0: lanes 0-15, 1: lanes 16-31
  int mn   = lane & 15;

  v8f acc[8] = {};
  if (ACC) {
    // C/D layout: VGPR v <-> row m0+v+8*half, col = lane%16 (per 16x16 tile)
    #pragma unroll
    for (int j = 0; j < 8; ++j)
      #pragma unroll
      for (int v = 0; v < 8; ++v)
        acc[j][v] = Cin[(size_t)(m0 + v + 8 * half) * CH + j * 16 + mn];
  }

  const float* Arow = A + (size_t)(m0 + mn) * CH;
  for (int k = 0; k < CH; k += 4) {
    // A 16x4 fragment: lanes 0-15 hold K={k,k+1}, lanes 16-31 hold K={k+2,k+3}
    v2f a;
    a.x = Arow[k + 2 * half];
    a.y = Arow[k + 2 * half + 1];
    int kb = (k >> 1) + half;              // (k + 2*half) / 2
    #pragma unroll
    for (int j = 0; j < 8; ++j) {
      float2 t = Bs[kb * CH + j * 16 + mn];   // single aligned ds_load_b64
      v2f b;
      b.x = t.x;
      b.y = t.y;
      acc[j] = __builtin_amdgcn_wmma_f32_16x16x4_f32(
          /*neg_a=*/false, a, /*neg_b=*/false, b,
          /*c_mod=*/(short)0, acc[j], /*reuse_a=*/false, /*reuse_b=*/false);
    }
  }

  #pragma unroll
  for (int j = 0; j < 8; ++j) {
    #pragma unroll
    for (int v = 0; v < 8; ++v) {
      int row = m0 + v + 8 * half;
      int col = j * 16 + mn;
      float x = acc[j][v];
      if (RESID) x += resid[(size_t)row * CH + col];
      if (RELU) x = fmaxf(x, 0.0f);
      if (BIAS) x += bias[col];
      D[(size_t)row * CH + col] = x;
    }
  }
}

// ---------------------------------------------------------------------------
// One RGCN layer:
//   for r: tmp = 0; tmp[dst] += h[src] (et==r); agg (+)= tmp @ W_r
//   out = h @ SW + agg (+h if residual) -> relu? -> + bias
// ---------------------------------------------------------------------------
static void run_layer(const float* h, const int* src, const int* dst,
                      const int* et, const float* W, const float* SW,
                      const float* bias, bool relu, bool residual, float* tmp,
                      float* agg, float* out, hipStream_t stream) {
  const int n4 = N_NODES * CH / 4;                       // 1.6M float4
  dim3 zgrid((n4 + 255) / 256);                          // 6250 blocks
  dim3 sgrid((N_EDGES * 32 + 255) / 256);                // 200000 blocks
  dim3 ggrid((N_NODES / 16 + 7) / 8);                    // 391 blocks

  for (int r = 0; r < NREL; ++r) {
    zero_f32<<<zgrid, 256, 0, stream>>>((float4*)tmp, n4);
    rgcn_scatter<<<sgrid, 256, 0, stream>>>(h, src, dst, et, r, tmp);
    const float* Wr = W + (size_t)r * CH * CH;
    if (r == 0)
      rgcn_gemm<false, false, false, false>
          <<<ggrid, 256, 0, stream>>>(tmp, Wr, nullptr, agg, nullptr, nullptr,
                                      N_NODES);
    else
      rgcn_gemm<true, false, false, false>
          <<<ggrid, 256, 0, stream>>>(tmp, Wr, agg, agg, nullptr, nullptr,
                                      N_NODES);
  }
  if (residual)  // layer 1: +h residual, relu, bias
    rgcn_gemm<true, true, true, true>
        <<<ggrid, 256, 0, stream>>>(h, SW, agg, out, h, bias, N_NODES);
  else           // layer 2: bias only
    rgcn_gemm<true, false, false, true>
        <<<ggrid, 256, 0, stream>>>(h, SW, agg, out, nullptr, bias, N_NODES);
  (void)relu;
}

// ---------------------------------------------------------------------------
extern "C" void kernel_launch(void* const* d_in, const int* in_sizes, int n_in,
                              void* d_out, int out_size, void* d_ws,
                              size_t ws_size, hipStream_t stream) {
  const float* x   = (const float*)d_in[0];
  const int*   ei  = (const int*)d_in[1];   // [2, E]
  const int*   et  = (const int*)d_in[2];   // [E]
  const float* w1  = (const float*)d_in[3]; // [R,128,128]
  const float* sw1 = (const float*)d_in[4]; // [128,128]
  const float* b1  = (const float*)d_in[5]; // [128]
  const float* w2  = (const float*)d_in[6];
  const float* sw2 = (const float*)d_in[7];
  const float* b2  = (const float*)d_in[8];
  float* out = (float*)d_out;

  const int* src = ei;
  const int* dst = ei + N_EDGES;

  // Workspace: tmp[N,C] + agg[N,C] + hmid[N,C] = 76.8 MB
  float* tmp  = (float*)d_ws;
  float* agg  = tmp + (size_t)N_NODES * CH;
  float* hmid = agg + (size_t)N_NODES * CH;

  // Layer 1: relu + identity residual + bias
  run_layer(x, src, dst, et, w1, sw1, b1, true, true, tmp, agg, hmid, stream);
  // Layer 2: plain + bias
  run_layer(hmid, src, dst, et, w2, sw2, b2, false, false, tmp, agg, out,
            stream);
}